// ProtoLoss_8718783610904
// MI455X (gfx1250) — compile-verified
//
#include <hip/hip_runtime.h>
#include <math.h>

typedef float v2f __attribute__((ext_vector_type(2)));
typedef float v8f __attribute__((ext_vector_type(8)));

constexpr int kN = 262144;
constexpr int kD = 128;
constexpr int kF = 64;

constexpr int NBLK1 = 256;                       // segment-sum blocks
constexpr int NBLK3 = 1024;                      // main-pass blocks
constexpr int WAVES3 = 8;                        // 256 threads / 32
constexpr int TILES  = kN / 16;                  // 16384 row-tiles
constexpr int TILES_PER_WAVE = TILES / (NBLK3 * WAVES3); // 2

// workspace layout (floats)
constexpr int WS_SUMS  = 0;            // [64*128] proto sums -> protos (in place)
constexpr int WS_CNT   = 8192;         // [64] counts
constexpr int WS_PNORM = 8256;         // [64] prototype norms
constexpr int WS_PART  = 8320;         // [NBLK3] per-block loss partials
constexpr int WS_TOTAL = WS_PART + NBLK3;

__global__ __launch_bounds__(256) void k0_zero(float* ws) {
    int i = blockIdx.x * 256 + threadIdx.x;
    if (i < WS_TOTAL) ws[i] = 0.0f;
}

// ---- pass 1: per-family sums & counts (LDS-accumulate, one global flush) ----
__global__ __launch_bounds__(256) void k1_segsum(const float* __restrict__ emb,
                                                 const int* __restrict__ fam,
                                                 float* __restrict__ sums,
                                                 float* __restrict__ counts) {
    __shared__ float lsum[kF * kD];
    __shared__ float lcnt[kF];
    const int tid = threadIdx.x;
    for (int i = tid; i < kF * kD; i += 256) lsum[i] = 0.0f;
    if (tid < kF) lcnt[tid] = 0.0f;
    __syncthreads();

    const int wave = tid >> 5, lane = tid & 31;
    const int rowsPerBlock = kN / NBLK1;          // 1024
    const int base = blockIdx.x * rowsPerBlock;
    for (int r = wave; r < rowsPerBlock; r += WAVES3) {
        const int row = base + r;                 // one wave handles one row
        const int f = fam[row];
        const float4 v = ((const float4*)(emb + (size_t)row * kD))[lane];
        float* dst = &lsum[f * kD + lane * 4];
        atomicAdd(dst + 0, v.x);
        atomicAdd(dst + 1, v.y);
        atomicAdd(dst + 2, v.z);
        atomicAdd(dst + 3, v.w);
        if (lane == 0) atomicAdd(&lcnt[f], 1.0f);
    }
    __syncthreads();
    for (int i = tid; i < kF * kD; i += 256) atomicAdd(&sums[i], lsum[i]);
    if (tid < kF) atomicAdd(&counts[tid], lcnt[tid]);
}

// ---- pass 2: protos = sums / max(count,1); p_norm ----
__global__ __launch_bounds__(256) void k2_final(float* __restrict__ sums,
                                                const float* __restrict__ counts,
                                                float* __restrict__ pnorm) {
    __shared__ float lp[kF * kD];
    __shared__ float lc[kF];
    const int tid = threadIdx.x;
    if (tid < kF) lc[tid] = fmaxf(counts[tid], 1.0f);
    __syncthreads();
    for (int i = tid; i < kF * kD; i += 256) {
        float p = sums[i] / lc[i >> 7];
        sums[i] = p;               // protos stored in place
        lp[i] = p;
    }
    __syncthreads();
    if (tid < kF) {
        float s = 0.0f;
        for (int d = 0; d < kD; ++d) { float p = lp[tid * kD + d]; s += p * p; }
        pnorm[tid] = sqrtf(s);
    }
}

// ---- pass 3: fused  sim = (E @ P^T)/(|e||p|)  ->  -2*pos + logsumexp(4*sim) ----
__global__ __launch_bounds__(256) void k3_main(const float* __restrict__ emb,
                                               const int* __restrict__ fam,
                                               const float* __restrict__ protos,
                                               const float* __restrict__ pnorm,
                                               float* __restrict__ partials) {
    __shared__ float ldsp[kF * 132];   // padded stride 132 -> conflict-free b64 reads
    __shared__ float ldspn[kF];
    __shared__ float wsum[WAVES3];

    const int tid = threadIdx.x;
    for (int i = tid; i < kF * kD; i += 256)
        ldsp[(i >> 7) * 132 + (i & 127)] = protos[i];
    if (tid < kF) ldspn[tid] = pnorm[tid];
    __syncthreads();

    const int wave = tid >> 5;
    const int lane = tid & 31;
    const int lo   = lane & 15;
    const int hi   = lane >> 4;        // 0 | 1
    const int kOff = hi * 2;           // A/B K-offset per 16x16x4 f32 layout

    float wacc = 0.0f;
    const int gw = blockIdx.x * WAVES3 + wave;

    for (int t = 0; t < TILES_PER_WAVE; ++t) {
        const int tile = gw * TILES_PER_WAVE + t;
        const int row0 = tile * 16;
        const float* arow = emb + (size_t)(row0 + lo) * kD;

        v8f z;
        #pragma unroll
        for (int i = 0; i < 8; ++i) z[i] = 0.0f;
        v8f acc0 = z, acc1 = z, acc2 = z, acc3 = z;

        float sumsq = 0.0f;

        #pragma unroll 4
        for (int k0 = 0; k0 < kD; k0 += 4) {
            v2f a = *(const v2f*)(arow + k0 + kOff);            // A frag (global b64)
            sumsq += a.x * a.x + a.y * a.y;
            v2f b0 = *(const v2f*)&ldsp[(0 * 16 + lo) * 132 + k0 + kOff];
            v2f b1 = *(const v2f*)&ldsp[(1 * 16 + lo) * 132 + k0 + kOff];
            v2f b2 = *(const v2f*)&ldsp[(2 * 16 + lo) * 132 + k0 + kOff];
            v2f b3 = *(const v2f*)&ldsp[(3 * 16 + lo) * 132 + k0 + kOff];
            acc0 = __builtin_amdgcn_wmma_f32_16x16x4_f32(false, a, false, b0, (short)0, acc0, false, false);
            acc1 = __builtin_amdgcn_wmma_f32_16x16x4_f32(false, a, false, b1, (short)0, acc1, false, false);
            acc2 = __builtin_amdgcn_wmma_f32_16x16x4_f32(false, a, false, b2, (short)0, acc2, false, false);
            acc3 = __builtin_amdgcn_wmma_f32_16x16x4_f32(false, a, false, b3, (short)0, acc3, false, false);
        }

        // full row norm: lane lo covers K%4 in {kOff,kOff+1}; pair with lane^16
        sumsq += __shfl_xor(sumsq, 16, 32);
        const float enorm = sqrtf(sumsq);          // norm of row (row0+lo)
        const int famv = fam[row0 + lo];           // family of row (row0+lo)

        // broadcast per-output-row norm/family: D VGPR j -> rows j (lanes<16), j+8
        float enj[8]; int fj[8];
        #pragma unroll
        for (int j = 0; j < 8; ++j) {
            const int rowj = j + 8 * hi;
            enj[j] = __shfl(enorm, rowj, 32);
            fj[j]  = __shfl(famv, rowj, 32);
        }

        float sumExp[8], posv[8];
        #pragma unroll
        for (int j = 0; j < 8; ++j) { sumExp[j] = 0.0f; posv[j] = 0.0f; }

        #pragma unroll
        for (int nt = 0; nt < 4; ++nt) {
            const int c = nt * 16 + lo;            // prototype column this lane holds
            const float pn = ldspn[c];
            const v8f accv = (nt == 0) ? acc0 : (nt == 1) ? acc1 : (nt == 2) ? acc2 : acc3;
            #pragma unroll
            for (int j = 0; j < 8; ++j) {
                const float denom = fmaxf(enj[j] * pn, 1e-8f);
                const float s = accv[j] / denom;   // cosine sim
                sumExp[j] += expf(4.0f * s);       // sim / T^2, T=0.5
                posv[j]   += (c == fj[j]) ? s : 0.0f;
            }
        }

        // reduce across the 16 columns held by each half-wave
        #pragma unroll
        for (int j = 0; j < 8; ++j) {
            float se = sumExp[j], pv = posv[j];
            #pragma unroll
            for (int m = 1; m < 16; m <<= 1) {
                se += __shfl_xor(se, m, 32);
                pv += __shfl_xor(pv, m, 32);
            }
            const float ps = -2.0f * pv + logf(se);   // -pos/T + lse
            wacc += (lo == 0) ? ps : 0.0f;            // lanes 0 & 16 contribute
        }
    }

    wacc += __shfl_xor(wacc, 16, 32);
    if (lane == 0) wsum[wave] = wacc;
    __syncthreads();
    if (tid == 0) {
        float s = 0.0f;
        for (int w = 0; w < WAVES3; ++w) s += wsum[w];   // fixed order
        partials[blockIdx.x] = s;
    }
}

// ---- pass 4: deterministic fixed-order final reduction ----
__global__ void k4_reduce(const float* __restrict__ partials, float* __restrict__ out) {
    if (threadIdx.x == 0 && blockIdx.x == 0) {
        float s = 0.0f;
        for (int i = 0; i < NBLK3; ++i) s += partials[i];
        out[0] = s / (float)kN;
    }
}

extern "C" void kernel_launch(void* const* d_in, const int* in_sizes, int n_in,
                              void* d_out, int out_size, void* d_ws, size_t ws_size,
                              hipStream_t stream) {
    const float* emb = (const float*)d_in[0];
    const int*   fam = (const int*)d_in[1];
    float* ws       = (float*)d_ws;
    float* sums     = ws + WS_SUMS;     // becomes protos after k2
    float* counts   = ws + WS_CNT;
    float* pnorm    = ws + WS_PNORM;
    float* partials = ws + WS_PART;
    float* out      = (float*)d_out;

    k0_zero  <<<(WS_TOTAL + 255) / 256, 256, 0, stream>>>(ws);
    k1_segsum<<<NBLK1, 256, 0, stream>>>(emb, fam, sums, counts);
    k2_final <<<1, 256, 0, stream>>>(sums, counts, pnorm);
    k3_main  <<<NBLK3, 256, 0, stream>>>(emb, fam, sums, pnorm, partials);
    k4_reduce<<<1, 32, 0, stream>>>(partials, out);
}